// HouseholderRoPE_1932735283217
// MI455X (gfx1250) — compile-verified
//
#include <hip/hip_runtime.h>
#include <math.h>

// Problem constants (B,H,T,D,M) = (2,16,8192,64,8)
#define BB 2
#define HH 16
#define TT 8192
#define DD 64
#define MM 8

typedef float v2f __attribute__((ext_vector_type(2)));
typedef float v8f __attribute__((ext_vector_type(8)));

// Swizzled W layout: per head, 16(j) * 4(n) * 32(lane) v2f entries.
// Entry[(j*4+n)*32 + l] = { W[4j+2*(l>>4)][n*16+(l&15)], W[4j+1+2*(l>>4)][n*16+(l&15)] }
// so each WMMA B fragment is ONE aligned ds_load_b64 per lane.
#define WSW_PER_HEAD (16 * 4 * 32)            // 2048 v2f = 16 KB
#define WSW_BYTES    (HH * WSW_PER_HEAD * 8)  // 256 KB
#define TAB_BYTES    ((size_t)TT * (DD / 2) * 8)  // 2 MB of {cos,sin}

// ---------------------------------------------------------------------------
// Kernel 1: collapse Householder stack into W_h (z_out = z @ W_h), store in
// WMMA-fragment-swizzled order. grid = H, block = 64.
// ---------------------------------------------------------------------------
__global__ void hh_build_W(const float* __restrict__ V, v2f* __restrict__ Wsw) {
    const int h   = blockIdx.x;
    const int tid = threadIdx.x;               // 0..63 -> row of W
    __shared__ float u[MM][DD];
    __shared__ float sW[DD][DD];

    if (tid < MM) {
        const float* v = V + (h * MM + tid) * DD;
        float s = 0.0f;
        #pragma unroll
        for (int d = 0; d < DD; ++d) { float x = v[d]; s += x * x; }
        const float inv = 1.0f / sqrtf(s + 1e-16f);   // EPS^2 = 1e-16
        #pragma unroll
        for (int d = 0; d < DD; ++d) u[tid][d] = v[d] * inv;
    }
    __syncthreads();

    float w[DD];
    #pragma unroll
    for (int d = 0; d < DD; ++d) w[d] = (d == tid) ? 1.0f : 0.0f;
    for (int m = MM - 1; m >= 0; --m) {        // reference's reverse order
        float proj = 0.0f;
        #pragma unroll
        for (int d = 0; d < DD; ++d) proj += w[d] * u[m][d];
        proj *= 2.0f;
        #pragma unroll
        for (int d = 0; d < DD; ++d) w[d] -= proj * u[m][d];
    }
    #pragma unroll
    for (int d = 0; d < DD; ++d) sW[tid][d] = w[d];
    __syncthreads();

    // Emit swizzled fragments.
    v2f* dst = Wsw + h * WSW_PER_HEAD;
    for (int idx = tid; idx < WSW_PER_HEAD; idx += 64) {
        const int l    = idx & 31;
        const int jn   = idx >> 5;
        const int j    = jn >> 2;
        const int n    = jn & 3;
        const int krow = 4 * j + 2 * (l >> 4);
        const int col  = n * 16 + (l & 15);
        v2f e; e.x = sW[krow][col]; e.y = sW[krow + 1][col];
        dst[idx] = e;
    }
}

// ---------------------------------------------------------------------------
// Kernel 1b: RoPE cos/sin table, ctab[t*32+p] = {cos, sin}(pos[t]*freq[p]).
// ---------------------------------------------------------------------------
__global__ void hh_build_tab(const float* __restrict__ pos,
                             const float* __restrict__ freq,
                             v2f* __restrict__ ctab) {
    const int i = blockIdx.x * 256 + threadIdx.x;  // < T*32
    const int t = i >> 5, p = i & 31;
    float s, c;
    sincosf(pos[t] * freq[p], &s, &c);
    v2f e; e.x = c; e.y = s;
    ctab[i] = e;
}

// ---------------------------------------------------------------------------
// Kernel 2: (rows @ W_h) via V_WMMA_F32_16X16X4_F32, then RoPE.
// block = 256 (8 waves, one 16-row tile each); grid = 2*B*H*(T/128) = 4096.
// ---------------------------------------------------------------------------
template <bool USE_TAB>
__global__ void __launch_bounds__(256)
hh_gemm_rope(const float* __restrict__ q,
             const float* __restrict__ k,
             const v2f* __restrict__ Wsw,
             const v2f* __restrict__ ctab,
             const float* __restrict__ pos,
             const float* __restrict__ freq,
             float* __restrict__ out) {
    int bid = blockIdx.x;
    const int chunk  = bid & 63;  bid >>= 6;   // T/128 = 64
    const int h      = bid & 15;  bid >>= 4;
    const int b      = bid & 1;   bid >>= 1;
    const int tensor = bid;                    // 0 = q, 1 = k

    const float* src   = tensor ? k : q;
    const size_t slice = ((size_t)b * HH + h) * (size_t)TT * DD;
    const float* base  = src + slice;
    float*       obase = out + (size_t)tensor * ((size_t)BB * HH * TT * DD) + slice;
    const int    t0    = chunk * 128;

    __shared__ v2f   sWs[WSW_PER_HEAD];  // 16 KB, fragment-swizzled W_h
    __shared__ float sC[8][16 * DD];     // 32 KB, per-wave C staging

    // Coalesced straight copy of the pre-swizzled W_h.
    {
        const v2f* whp = Wsw + h * WSW_PER_HEAD;
        for (int i = threadIdx.x; i < WSW_PER_HEAD; i += 256) sWs[i] = whp[i];
    }
    __syncthreads();

    const int lane  = threadIdx.x & 31;
    const int wave  = threadIdx.x >> 5;
    const int col   = lane & 15;
    const int khalf = lane >> 4;         // K sub-pair select
    const int tbase = t0 + wave * 16;

    // A: lane reads K pair {4j+2*khalf, +1} of row (tbase+col) per k-step.
    const float* arow = base + (size_t)(tbase + col) * DD + 2 * khalf;

    v8f acc[4];
    acc[0] = v8f{}; acc[1] = v8f{}; acc[2] = v8f{}; acc[3] = v8f{};

    #pragma unroll
    for (int j = 0; j < 16; ++j) {                 // K = 64 in steps of 4
        const v2f a = *(const v2f*)(arow + 4 * j); // global_load_b64
        #pragma unroll
        for (int n = 0; n < 4; ++n) {              // N = 64 in 4 tiles
            const v2f bf = sWs[(j * 4 + n) * 32 + lane];  // ds_load_b64
            acc[n] = __builtin_amdgcn_wmma_f32_16x16x4_f32(
                false, a, false, bf, (short)0, acc[n], false, false);
        }
    }

    // Stage C: lanes 0-15 -> M=vgpr, lanes 16-31 -> M=vgpr+8 (ISA layout).
    float* cst = sC[wave];
    #pragma unroll
    for (int v = 0; v < 8; ++v) {
        const int m = v + 8 * khalf;
        #pragma unroll
        for (int n = 0; n < 4; ++n)
            cst[m * DD + n * 16 + col] = acc[n][v];
    }
    __syncthreads();

    // RoPE: iteration = row, lane = pair index p (32 pairs per row).
    #pragma unroll 4
    for (int it = 0; it < 16; ++it) {
        const int t = tbase + it;
        float c, s;
        if (USE_TAB) {
            const v2f cs = ctab[(size_t)t * 32 + lane];  // coalesced b64
            c = cs.x; s = cs.y;
        } else {
            sincosf(pos[t] * freq[lane], &s, &c);
        }
        const v2f x = *(const v2f*)(cst + it * DD + 2 * lane);
        v2f r;
        r.x = x.x * c - x.y * s;
        r.y = x.x * s + x.y * c;
        *(v2f*)(obase + (size_t)t * DD + 2 * lane) = r;
    }
}

// ---------------------------------------------------------------------------
extern "C" void kernel_launch(void* const* d_in, const int* in_sizes, int n_in,
                              void* d_out, int out_size, void* d_ws, size_t ws_size,
                              hipStream_t stream) {
    const float* q    = (const float*)d_in[0];
    const float* k    = (const float*)d_in[1];
    const float* V    = (const float*)d_in[2];
    const float* pos  = (const float*)d_in[3];
    const float* freq = (const float*)d_in[4];
    float*       out  = (float*)d_out;

    v2f* Wsw  = (v2f*)d_ws;                              // 256 KB
    v2f* ctab = (v2f*)((char*)d_ws + WSW_BYTES);         // +2 MB

    hh_build_W<<<HH, 64, 0, stream>>>(V, Wsw);

    const int nblocks = 2 * BB * HH * (TT / 128);        // 4096
    if (ws_size >= WSW_BYTES + TAB_BYTES) {
        hh_build_tab<<<(TT * 32) / 256, 256, 0, stream>>>(pos, freq, ctab);
        hh_gemm_rope<true><<<nblocks, 256, 0, stream>>>(q, k, Wsw, ctab, pos, freq, out);
    } else {
        hh_gemm_rope<false><<<nblocks, 256, 0, stream>>>(q, k, Wsw, ctab, pos, freq, out);
    }
}